// TopKRouter_42159398977857
// MI455X (gfx1250) — compile-verified
//
#include <hip/hip_runtime.h>
#include <hip/hip_bf16.h>

typedef __attribute__((ext_vector_type(2))) float v2f;
typedef __attribute__((ext_vector_type(4))) float v4f;
typedef __attribute__((ext_vector_type(8))) float v8f;
typedef __attribute__((ext_vector_type(4))) int   v4i;

typedef __attribute__((address_space(1))) v4i gv4i;   // global int4
typedef __attribute__((address_space(3))) v4i lv4i;   // LDS int4

#define DIM     4096
#define NEXP    64
#define NTOK    16384      // B*T = 4*4096
#define TILE_M  128
#define KC      32
#define LDSS    36         // padded stride (floats): 16B-aligned float4, conflict-free frags
#define NCHUNK  (DIM / KC) // 128

#if defined(__HIP_DEVICE_COMPILE__) && __has_builtin(__builtin_amdgcn_global_load_async_to_lds_b128)
#define HAVE_ASYNC 1
#else
#define HAVE_ASYNC 0
#endif

__device__ __forceinline__ void async_wait0() {
#if HAVE_ASYNC
#if __has_builtin(__builtin_amdgcn_s_wait_asynccnt)
  __builtin_amdgcn_s_wait_asynccnt(0);
#else
  asm volatile("s_wait_asynccnt 0" ::: "memory");
#endif
#endif
}

// Stage x tile: 128 rows x 8 float4 (1024 b128, 4 per thread). NT: pure stream.
__device__ __forceinline__ void stage_x(const float* __restrict__ x, float* lds,
                                        int tid, int mBase, int k0) {
#pragma unroll
  for (int i = 0; i < 4; ++i) {
    const int f   = tid + 256 * i;
    const int row = f >> 3;
    const int c4  = f & 7;
    const float* g = x + (size_t)(mBase + row) * DIM + k0 + c4 * 4;
    float*       l = lds + row * LDSS + c4 * 4;
#if HAVE_ASYNC
    __builtin_amdgcn_global_load_async_to_lds_b128(
        (gv4i*)g, (lv4i*)l, 0, /*cpol TH_NT*/ 1);
#else
    *(v4f*)l = __builtin_nontemporal_load((const v4f*)g);
#endif
  }
}

// Stage W tile: 64 rows x 8 float4 (512 b128, 2 per thread). RT: reused by all blocks.
__device__ __forceinline__ void stage_w(const float* __restrict__ W, float* lds,
                                        int tid, int k0) {
#pragma unroll
  for (int i = 0; i < 2; ++i) {
    const int f   = tid + 256 * i;
    const int row = f >> 3;
    const int c4  = f & 7;
    const float* g = W + (size_t)row * DIM + k0 + c4 * 4;
    float*       l = lds + row * LDSS + c4 * 4;
#if HAVE_ASYNC
    __builtin_amdgcn_global_load_async_to_lds_b128(
        (gv4i*)g, (lv4i*)l, 0, /*cpol RT*/ 0);
#else
    *(v4f*)l = *(const v4f*)g;
#endif
  }
}

// ---------------------------------------------------------------------------
// GEMM: logits[16384,64] = x[16384,4096] @ W[64,4096]^T  via v_wmma_f32_16x16x4_f32
// Double-buffered LDS; async global->LDS staging overlapped with WMMA compute.
// ---------------------------------------------------------------------------
__launch_bounds__(256, 2)
__global__ void router_gemm_kernel(const float* __restrict__ x,
                                   const float* __restrict__ W,
                                   float* __restrict__ logits) {
  __shared__ float lds_x[2][TILE_M * LDSS];
  __shared__ float lds_w[2][NEXP * LDSS];

  const int tid   = threadIdx.x;
  const int wave  = tid >> 5;
  const int lane  = tid & 31;
  const int lh    = lane & 15;          // M (A) / N (B) within 16
  const int kh    = (lane >> 4) << 1;   // K offset 0 or 2 for upper half-wave
  const int mBase = blockIdx.x * TILE_M;
  const int m0w   = wave * 16;

  v8f acc0 = {}, acc1 = {}, acc2 = {}, acc3 = {};

  // prologue: fill buffer 0
  stage_x(x, lds_x[0], tid, mBase, 0);
  stage_w(W, lds_w[0], tid, 0);
  async_wait0();
  __syncthreads();

  for (int c = 0; c < NCHUNK; ++c) {
    const int cur = c & 1;
    const int nxt = cur ^ 1;
    if (c + 1 < NCHUNK) {           // prefetch next chunk into the other buffer
      stage_x(x, lds_x[nxt], tid, mBase, (c + 1) * KC);
      stage_w(W, lds_w[nxt], tid, (c + 1) * KC);
    }

    // WMMA over current chunk: 8 steps of K=4, 4 N-tiles each
    const float* ax = &lds_x[cur][(m0w + lh) * LDSS + kh];
    const float* bw = &lds_w[cur][lh * LDSS + kh];
#pragma unroll
    for (int kk = 0; kk < KC; kk += 4) {
      v2f a  = *(const v2f*)(ax + kk);
      v2f b0 = *(const v2f*)(bw + 0 * 16 * LDSS + kk);
      v2f b1 = *(const v2f*)(bw + 1 * 16 * LDSS + kk);
      v2f b2 = *(const v2f*)(bw + 2 * 16 * LDSS + kk);
      v2f b3 = *(const v2f*)(bw + 3 * 16 * LDSS + kk);
      acc0 = __builtin_amdgcn_wmma_f32_16x16x4_f32(false, a, false, b0, (short)0, acc0, false, false);
      acc1 = __builtin_amdgcn_wmma_f32_16x16x4_f32(false, a, false, b1, (short)0, acc1, false, false);
      acc2 = __builtin_amdgcn_wmma_f32_16x16x4_f32(false, a, false, b2, (short)0, acc2, false, false);
      acc3 = __builtin_amdgcn_wmma_f32_16x16x4_f32(false, a, false, b3, (short)0, acc3, false, false);
    }

    async_wait0();      // my prefetch landed in LDS
    __syncthreads();    // everyone done reading cur + done landing nxt
  }

  // store C: lane l, vgpr g -> row = g + 8*(l/16), col = 16*t + l%16
  const int rowOut = mBase + m0w + ((lane >> 4) << 3);
#pragma unroll
  for (int g = 0; g < 8; ++g) {
    float* rp = logits + (size_t)(rowOut + g) * NEXP + lh;
    rp[0]  = acc0[g];
    rp[16] = acc1[g];
    rp[32] = acc2[g];
    rp[48] = acc3[g];
  }
}

// ---------------------------------------------------------------------------
// Top-2 + softmax over 64 experts, one thread per token (logits L2-resident)
// ---------------------------------------------------------------------------
__global__ void topk_softmax_kernel(const float* __restrict__ logits,
                                    int* __restrict__ out_idx,
                                    float* __restrict__ out_w) {
  const int r = blockIdx.x * blockDim.x + threadIdx.x;
  if (r >= NTOK) return;
  const v4f* row = (const v4f*)(logits + (size_t)r * NEXP);
  float b1 = -__builtin_inff(), b2 = -__builtin_inff();
  int i1 = 0, i2 = 0;
#pragma unroll
  for (int q = 0; q < NEXP / 4; ++q) {
    v4f v = row[q];
#pragma unroll
    for (int e = 0; e < 4; ++e) {
      const float val = v[e];
      const int   j   = q * 4 + e;
      if (val > b1)      { b2 = b1; i2 = i1; b1 = val; i1 = j; }
      else if (val > b2) { b2 = val; i2 = j; }
    }
  }
  const float e2    = __expf(b2 - b1);
  const float denom = 1.0f + e2;
  out_idx[r * 2 + 0] = i1;
  out_idx[r * 2 + 1] = i2;
  out_w[r * 2 + 0]   = 1.0f / denom;
  out_w[r * 2 + 1]   = e2 / denom;
}

// ---------------------------------------------------------------------------
extern "C" void kernel_launch(void* const* d_in, const int* in_sizes, int n_in,
                              void* d_out, int out_size, void* d_ws, size_t ws_size,
                              hipStream_t stream) {
  const float* x = (const float*)d_in[0];   // [4,4096,4096] f32
  const float* W = (const float*)d_in[1];   // [64,4096] f32

  // d_out layout (return order): topk_idx [NTOK*2] i32, topk_weight [NTOK*2] f32,
  // router_logits [NTOK*64] f32
  int*   out_idx    = (int*)d_out;
  float* out_w      = (float*)d_out + NTOK * 2;
  float* out_logits = (float*)d_out + NTOK * 4;

  router_gemm_kernel<<<NTOK / TILE_M, 256, 0, stream>>>(x, W, out_logits);
  topk_softmax_kernel<<<NTOK / 256, 256, 0, stream>>>(out_logits, out_idx, out_w);
}